// ReconDistanceLoss_19645180411971
// MI455X (gfx1250) — compile-verified
//
#include <hip/hip_runtime.h>
#include <math.h>

typedef __attribute__((ext_vector_type(2))) float v2f;
typedef __attribute__((ext_vector_type(8))) float v8f;

#define RL_EPS 1e-7f
#define MAN_BLOCKS 64

// ---------------------------------------------------------------------------
// Kernel 0: build augmented pc array  aug[m] = (x, y, z, x^2+y^2+z^2)
// so each lane's B-operand pair is one aligned b64 load (group0: (x,y),
// group1: (z, ||y||^2)) -- removes all B-construction VALU from the hot loop.
// ---------------------------------------------------------------------------
__global__ __launch_bounds__(256) void rl_prep_aug(const float* __restrict__ pc,
                                                   int M, float* __restrict__ aug) {
    int m = blockIdx.x * blockDim.x + threadIdx.x;
    if (m < M) {
        const float x = pc[3 * m], y = pc[3 * m + 1], z = pc[3 * m + 2];
        aug[4 * m + 0] = x;
        aug[4 * m + 1] = y;
        aug[4 * m + 2] = z;
        aug[4 * m + 3] = x * x + y * y + z * z;
    }
}

// ---------------------------------------------------------------------------
// Kernel 1: sum |manifold_pnts_pred| -> one partial per block (wave-sized blocks)
// ---------------------------------------------------------------------------
__global__ __launch_bounds__(32) void rl_manifold_partial(const float* __restrict__ pred,
                                                          int n, float* __restrict__ mp) {
    float s = 0.0f;
    for (int i = blockIdx.x * 32 + threadIdx.x; i < n; i += gridDim.x * 32)
        s += fabsf(pred[i]);
    s += __shfl_xor(s, 16, 32);
    s += __shfl_xor(s, 8, 32);
    s += __shfl_xor(s, 4, 32);
    s += __shfl_xor(s, 2, 32);
    s += __shfl_xor(s, 1, 32);
    if (threadIdx.x == 0) mp[blockIdx.x] = s;
}

// ---------------------------------------------------------------------------
// Shared epilogue: cross-lane min, per-row first-term, per-wave partial store.
// ---------------------------------------------------------------------------
__device__ __forceinline__ void rl_epilogue(v8f acc, int n0, int half, int lane,
                                            int wave, int waves, int Nz, int Ng,
                                            const float* __restrict__ zls,
                                            const float* __restrict__ gls,
                                            const float* __restrict__ zev,
                                            const float* __restrict__ gev,
                                            float* __restrict__ fp) {
    const int N = Nz + Ng;
#pragma unroll
    for (int v = 0; v < 8; ++v) {
        float m = acc[v];
        m = fminf(m, __shfl_xor(m, 1, 32));
        m = fminf(m, __shfl_xor(m, 2, 32));
        m = fminf(m, __shfl_xor(m, 4, 32));
        m = fminf(m, __shfl_xor(m, 8, 32));
        acc[v] = m;   // rowmin for proj row  n0 + v + 8*half
    }
    float s = 0.0f;
    const int base = n0 + 8 * half;
#pragma unroll
    for (int v = 0; v < 8; ++v) {
        const int n = base + v;
        if (n < N) {
            const float* sp = (n < Nz) ? (zls + 3 * n) : (gls + 3 * (n - Nz));
            const float x0 = sp[0], x1 = sp[1], x2c = sp[2];
            const float dmin = acc[v] + x0 * x0 + x1 * x1 + x2c * x2c;
            const float ev = (n < Nz) ? zev[n] : gev[n - Nz];
            s += fabsf(sqrtf(fabsf(dmin) + RL_EPS) - fabsf(ev));
        }
    }
    s += __shfl_xor(s, 16, 32);   // combine the two halves -> full 16-row sum
    if (lane == 0 && wave < waves) fp[wave] = s;
}

// ---------------------------------------------------------------------------
// A-operand setup (shared): row n0+mrow, augmented (-2x0,-2x1,-2x2,1)
// ---------------------------------------------------------------------------
__device__ __forceinline__ v2f rl_make_a(int n0, int mrow, int half, int Nz, int Ng,
                                         const float* __restrict__ zls,
                                         const float* __restrict__ gls) {
    const int N = Nz + Ng;
    int nr = n0 + mrow; nr = (nr < N) ? nr : (N - 1);
    const float* ps = (nr < Nz) ? (zls + 3 * nr) : (gls + 3 * (nr - Nz));
    const float px = ps[0], py = ps[1], pz = ps[2];
    v2f a;
    a.x = half ? (-2.0f * pz) : (-2.0f * px);   // VGPR0: K=0 (lo) / K=2 (hi)
    a.y = half ? 1.0f         : (-2.0f * py);   // VGPR1: K=1 (lo) / K=3 (hi)
    return a;
}

// ---------------------------------------------------------------------------
// Kernel 2a (fast): WMMA sweep over pre-augmented pc. Full tiles use purely
// affine addressing (pointer bump, 256 B/tile) so the unrolled clause gets
// immediate-offset b64 loads; the (rare) remainder tile sits behind a
// wave-uniform scalar branch so EXEC stays all-ones for every WMMA.
// ---------------------------------------------------------------------------
__global__ __launch_bounds__(256) void rl_min_dist_aug(const float* __restrict__ zls,
                                                       const float* __restrict__ gls,
                                                       const float* __restrict__ aug,
                                                       const float* __restrict__ zev,
                                                       const float* __restrict__ gev,
                                                       int Nz, int Ng, int M, int waves,
                                                       float* __restrict__ fp) {
    const int lane = threadIdx.x & 31;
    const int wave = (blockIdx.x * blockDim.x + threadIdx.x) >> 5;
    const int w = (wave < waves) ? wave : (waves - 1);   // clamp, keep EXEC full
    const int n0 = w * 16;
    const int half = lane >> 4;
    const int mrow = lane & 15;

    const v2f a = rl_make_a(n0, mrow, half, Nz, Ng, zls, gls);

    v8f acc;
#pragma unroll
    for (int v = 0; v < 8; ++v) acc[v] = 3.0e38f;

    const v8f czero = {};
    // per-lane B source: aug + 4*col + 2*half  (8-byte aligned pairs)
    const float* p = aug + 2 * half + 4 * mrow;
    const int tilesFull = M >> 4;            // affine full tiles
    for (int t = 0; t < tilesFull; ++t) {
        const v2f b = *(const v2f*)(p);
        p += 64;                             // 16 cols * 4 floats
        v8f d = __builtin_amdgcn_wmma_f32_16x16x4_f32(
            false, a, false, b, (short)0, czero, false, false);
#pragma unroll
        for (int v = 0; v < 8; ++v) acc[v] = fminf(acc[v], d[v]);
    }
    if (M & 15) {                            // wave-uniform remainder tile
        int col = tilesFull * 16 + mrow;     // duplicate-clamp: min-safe
        col = (col < M) ? col : (M - 1);
        const v2f b = *(const v2f*)(aug + 4 * col + 2 * half);
        v8f d = __builtin_amdgcn_wmma_f32_16x16x4_f32(
            false, a, false, b, (short)0, czero, false, false);
#pragma unroll
        for (int v = 0; v < 8; ++v) acc[v] = fminf(acc[v], d[v]);
    }
    rl_epilogue(acc, n0, half, lane, wave, waves, Nz, Ng, zls, gls, zev, gev, fp);
}

// ---------------------------------------------------------------------------
// Kernel 2b (fallback, no workspace for aug): on-the-fly B construction.
// ---------------------------------------------------------------------------
__global__ __launch_bounds__(256) void rl_min_dist(const float* __restrict__ zls,
                                                   const float* __restrict__ gls,
                                                   const float* __restrict__ pc,
                                                   const float* __restrict__ zev,
                                                   const float* __restrict__ gev,
                                                   int Nz, int Ng, int M, int waves,
                                                   float* __restrict__ fp) {
    const int lane = threadIdx.x & 31;
    const int wave = (blockIdx.x * blockDim.x + threadIdx.x) >> 5;
    const int w = (wave < waves) ? wave : (waves - 1);
    const int n0 = w * 16;
    const int half = lane >> 4;
    const int mrow = lane & 15;

    const v2f a = rl_make_a(n0, mrow, half, Nz, Ng, zls, gls);

    v8f acc;
#pragma unroll
    for (int v = 0; v < 8; ++v) acc[v] = 3.0e38f;

    const v8f czero = {};
    const int tiles = (M + 15) >> 4;
    for (int t = 0; t < tiles; ++t) {
        int col = t * 16 + mrow; col = (col < M) ? col : (M - 1);
        const float qx = pc[3 * col], qy = pc[3 * col + 1], qz = pc[3 * col + 2];
        const float q2 = qx * qx + qy * qy + qz * qz;
        v2f b;
        b.x = half ? qz : qx;
        b.y = half ? q2 : qy;
        v8f d = __builtin_amdgcn_wmma_f32_16x16x4_f32(
            false, a, false, b, (short)0, czero, false, false);
#pragma unroll
        for (int v = 0; v < 8; ++v) acc[v] = fminf(acc[v], d[v]);
    }
    rl_epilogue(acc, n0, half, lane, wave, waves, Nz, Ng, zls, gls, zev, gev, fp);
}

// ---------------------------------------------------------------------------
// Kernel 3: deterministic final reduction + loss assembly (3 floats out)
// ---------------------------------------------------------------------------
__global__ void rl_finalize(const float* __restrict__ fp, int waves,
                            const float* __restrict__ mp, int mblocks,
                            const int* __restrict__ lam, int N, int Nman,
                            float* __restrict__ out) {
    float fs = 0.0f;
    for (int i = 0; i < waves; ++i) fs += fp[i];
    float ms = 0.0f;
    for (int i = 0; i < mblocks; ++i) ms += mp[i];
    const float first_mean = fs / (float)N;
    const float second_mean = ms / (float)Nman;
    const float l = (float)lam[0];
    out[0] = first_mean + l * second_mean;
    out[1] = first_mean;
    out[2] = second_mean;
}

// ---------------------------------------------------------------------------
extern "C" void kernel_launch(void* const* d_in, const int* in_sizes, int n_in,
                              void* d_out, int out_size, void* d_ws, size_t ws_size,
                              hipStream_t stream) {
    (void)n_in; (void)out_size;
    const float* zls = (const float*)d_in[0];   // [8192,3]
    const float* gls = (const float*)d_in[1];   // [8192,3]
    const float* pc  = (const float*)d_in[2];   // [16384,3]
    const float* zev = (const float*)d_in[3];   // [8192,1]
    const float* gev = (const float*)d_in[4];   // [8192,1]
    const float* man = (const float*)d_in[5];   // [16384,1]
    const int*   lam = (const int*)d_in[6];     // scalar int

    const int Nz = in_sizes[0] / 3;
    const int Ng = in_sizes[1] / 3;
    const int M  = in_sizes[2] / 3;
    const int Nman = in_sizes[5];
    const int N = Nz + Ng;

    const int waves  = (N + 15) / 16;           // one wave per 16 proj rows
    const int blocks = (waves + 7) / 8;         // 8 wave32 per 256-thread block

    const size_t augBytes = (size_t)16 * (size_t)M;          // float4 per pc point
    const size_t redBytes = (size_t)4 * (size_t)(waves + MAN_BLOCKS);

    if (ws_size >= augBytes + redBytes) {
        float* aug = (float*)d_ws;              // [M][4] augmented pc
        float* fp  = aug + 4 * (size_t)M;       // [waves] per-wave partials
        float* mp  = fp + waves;                // [MAN_BLOCKS] manifold partials
        rl_prep_aug<<<(M + 255) / 256, 256, 0, stream>>>(pc, M, aug);
        rl_manifold_partial<<<MAN_BLOCKS, 32, 0, stream>>>(man, Nman, mp);
        rl_min_dist_aug<<<blocks, 256, 0, stream>>>(zls, gls, aug, zev, gev,
                                                    Nz, Ng, M, waves, fp);
        rl_finalize<<<1, 1, 0, stream>>>(fp, waves, mp, MAN_BLOCKS, lam, N, Nman,
                                         (float*)d_out);
    } else {
        float* fp = (float*)d_ws;
        float* mp = fp + waves;
        rl_manifold_partial<<<MAN_BLOCKS, 32, 0, stream>>>(man, Nman, mp);
        rl_min_dist<<<blocks, 256, 0, stream>>>(zls, gls, pc, zev, gev,
                                                Nz, Ng, M, waves, fp);
        rl_finalize<<<1, 1, 0, stream>>>(fp, waves, mp, MAN_BLOCKS, lam, N, Nman,
                                         (float*)d_out);
    }
}